// LinearGraphClassifier_20040317403820
// MI455X (gfx1250) — compile-verified
//
#include <hip/hip_runtime.h>
#include <hip/hip_bf16.h>

typedef __attribute__((ext_vector_type(2))) float v2f;
typedef __attribute__((ext_vector_type(8))) float v8f;

#define TK_D 1024      // feature dim
#define TK_C 8         // classes
#define TK_N 8192      // nodes per graph
#define TK_B 32        // graphs
#define TK_T (TK_B * TK_N)
#define TK_K (TK_N / 2)

// ---------------------------------------------------------------------------
// Pad lin_w [8,1024] -> wpad [16,1024] (rows 8..15 zero) so the WMMA B-matrix
// loads are branch-free.
// ---------------------------------------------------------------------------
__global__ void tk_wpad_kernel(const float* __restrict__ lin_w,
                               float* __restrict__ wpad) {
    int i = blockIdx.x * blockDim.x + threadIdx.x;   // 16*1024 total
    if (i < 16 * TK_D) {
        int r = i >> 10;                              // row
        wpad[i] = (r < TK_C) ? lin_w[i] : 0.0f;
    }
}

// ---------------------------------------------------------------------------
// node_pred = x @ wpad^T + b via V_WMMA_F32_16X16X4_F32.
// One wave computes a 16-row x 16-col (8 valid) tile, looping K=1024 in
// steps of 4. A-layout: lane<16 holds (M=lane, K=kb..kb+1), lane>=16 holds
// (M=lane-16, K=kb+2..kb+3). B-layout mirrors with N=lane&15.
// ---------------------------------------------------------------------------
__global__ void tk_gemm_kernel(const float* __restrict__ x,
                               const float* __restrict__ wpad,
                               const float* __restrict__ lin_b,
                               float* __restrict__ pred) {
    const int wave = threadIdx.x >> 5;      // 8 waves / block
    const int lane = threadIdx.x & 31;
    const int half = lane >> 4;             // 0 or 1
    const int ln   = lane & 15;
    const long rowBase = ((long)blockIdx.x * 8 + wave) * 16;

    const float* ap = x    + (rowBase + ln) * TK_D + 2 * half;  // A operand
    const float* bp = wpad + (long)ln * TK_D       + 2 * half;  // B operand

    v8f acc = {};
#pragma unroll 8
    for (int kb = 0; kb < TK_D; kb += 4) {
        v2f a = *(const v2f*)(ap + kb);
        v2f b = *(const v2f*)(bp + kb);
        acc = __builtin_amdgcn_wmma_f32_16x16x4_f32(
            /*neg_a=*/false, a, /*neg_b=*/false, b,
            /*c_mod=*/(short)0, acc, /*reuse_a=*/false, /*reuse_b=*/false);
    }

    // C/D layout: VGPR r -> M=r (lanes 0-15) / M=r+8 (lanes 16-31), N=ln.
    if (ln < TK_C) {
        float bias = lin_b[ln];
#pragma unroll
        for (int r = 0; r < 8; ++r) {
            long row = rowBase + r + 8 * half;
            pred[row * TK_C + ln] = acc[r] + bias;
        }
    }
}

// ---------------------------------------------------------------------------
// score = tanh(pred . w / ||w||); also emit order-preserving u32 key.
// ---------------------------------------------------------------------------
__global__ void tk_score_kernel(const float* __restrict__ pred,
                                const float* __restrict__ topk_w,
                                float* __restrict__ score,
                                unsigned* __restrict__ keys) {
    int t = blockIdx.x * blockDim.x + threadIdx.x;
    if (t >= TK_T) return;
    float nrm2 = 0.0f, dot = 0.0f;
#pragma unroll
    for (int c = 0; c < TK_C; ++c) {
        float w = topk_w[c];
        nrm2 += w * w;
        dot  += pred[(long)t * TK_C + c] * w;
    }
    float s = tanhf(dot * rsqrtf(nrm2));
    score[t] = s;
    unsigned u = __float_as_uint(s);
    keys[t] = (u & 0x80000000u) ? ~u : (u | 0x80000000u);  // monotonic map
}

// ---------------------------------------------------------------------------
// Exact K-th largest key per graph via 4-round MSB radix select.
// One 256-thread block per graph. Outputs kth key and R = #ties to include.
// ---------------------------------------------------------------------------
__global__ void tk_select_kernel(const unsigned* __restrict__ keys,
                                 unsigned* __restrict__ kthKey,
                                 int* __restrict__ tiesNeeded) {
    __shared__ int hist[256];
    __shared__ unsigned sPrefix;
    __shared__ int sK;
    const int b = blockIdx.x;
    const unsigned* gk = keys + (long)b * TK_N;

    if (threadIdx.x == 0) { sPrefix = 0u; sK = TK_K; }
    __syncthreads();

    for (int pass = 0; pass < 4; ++pass) {
        const int shift = 24 - 8 * pass;
        hist[threadIdx.x] = 0;
        __syncthreads();
        unsigned prefix = sPrefix;
        for (int i = threadIdx.x; i < TK_N; i += 256) {
            unsigned key = gk[i];
            bool cand = (pass == 0) || ((key >> (shift + 8)) == prefix);
            if (cand) atomicAdd(&hist[(key >> shift) & 255], 1);
        }
        __syncthreads();
        if (threadIdx.x == 0) {
            int k = sK, cum = 0, bin = 0;
            for (int j = 255; j >= 0; --j) {
                if (cum + hist[j] >= k) { bin = j; break; }
                cum += hist[j];
            }
            sPrefix = (prefix << 8) | (unsigned)bin;
            sK = k - cum;   // rank within the chosen bin
        }
        __syncthreads();
    }
    if (threadIdx.x == 0) {
        kthKey[b]     = sPrefix;      // exact K-th largest key
        tiesNeeded[b] = sK;           // include first sK ties (by index)
    }
}

// ---------------------------------------------------------------------------
// x_final[b] = (1/K) * sum over selected nodes of score * pred.
// Selected = key > kth, plus first R ties in node-index order (matches
// jax.lax.top_k's lowest-index tie-break). Deterministic: index-ordered tie
// ranking via block exclusive scan; fixed-order LDS tree reduction.
// ---------------------------------------------------------------------------
__global__ void tk_pool_kernel(const float* __restrict__ pred,
                               const float* __restrict__ score,
                               const unsigned* __restrict__ keys,
                               const unsigned* __restrict__ kthKey,
                               const int* __restrict__ tiesNeeded,
                               float* __restrict__ out) {
    const int PER = TK_N / 256;          // 32 contiguous nodes per thread
    __shared__ int scan[256];
    __shared__ float red[256][9];        // pad to 9 to dodge bank conflicts
    const int b = blockIdx.x, t = threadIdx.x;
    const long base = (long)b * TK_N;
    const unsigned kth = kthKey[b];
    const int R = tiesNeeded[b];

    // 1) count ties in this thread's index-contiguous chunk
    int lt = 0;
    for (int j = 0; j < PER; ++j)
        lt += (keys[base + (long)t * PER + j] == kth);
    scan[t] = lt;
    __syncthreads();
    // Hillis-Steele inclusive scan over 256 ints
    for (int off = 1; off < 256; off <<= 1) {
        int v = (t >= off) ? scan[t - off] : 0;
        __syncthreads();
        scan[t] += v;
        __syncthreads();
    }
    int tieRank = scan[t] - lt;          // exclusive prefix = global tie rank

    // 2) accumulate selected nodes
    float acc[TK_C] = {0,0,0,0,0,0,0,0};
    for (int j = 0; j < PER; ++j) {
        long i = base + (long)t * PER + j;
        unsigned key = keys[i];
        bool inc = false;
        if (key > kth) inc = true;
        else if (key == kth) { if (tieRank < R) inc = true; ++tieRank; }
        if (inc) {
            float s = score[i];
#pragma unroll
            for (int c = 0; c < TK_C; ++c) acc[c] += s * pred[i * TK_C + c];
        }
    }
#pragma unroll
    for (int c = 0; c < TK_C; ++c) red[t][c] = acc[c];
    __syncthreads();
    for (int off = 128; off > 0; off >>= 1) {
        if (t < off)
#pragma unroll
            for (int c = 0; c < TK_C; ++c) red[t][c] += red[t + off][c];
        __syncthreads();
    }
    if (t < TK_C) out[b * TK_C + t] = red[0][t] * (1.0f / (float)TK_K);
}

// ---------------------------------------------------------------------------
extern "C" void kernel_launch(void* const* d_in, const int* in_sizes, int n_in,
                              void* d_out, int out_size, void* d_ws, size_t ws_size,
                              hipStream_t stream) {
    (void)in_sizes; (void)n_in; (void)out_size; (void)ws_size;
    const float* x      = (const float*)d_in[0];   // [T, D]
    const float* lin_w  = (const float*)d_in[1];   // [C, D]
    const float* lin_b  = (const float*)d_in[2];   // [C]
    const float* topk_w = (const float*)d_in[3];   // [C]
    // d_in[4] edge_index, d_in[5] batch: unused by the reference math.

    float* out     = (float*)d_out;
    float* x_final = out;              // [B*C] = 256 floats
    float* pred    = out + TK_B * TK_C; // [T*C]

    // workspace layout (~2.2 MB)
    float*    score = (float*)d_ws;                       // T floats
    unsigned* keys  = (unsigned*)(score + TK_T);          // T u32
    float*    wpad  = (float*)(keys + TK_T);              // 16*1024 floats
    unsigned* kth   = (unsigned*)(wpad + 16 * TK_D);      // B u32
    int*      ties  = (int*)(kth + TK_B);                 // B int

    tk_wpad_kernel<<<(16 * TK_D + 255) / 256, 256, 0, stream>>>(lin_w, wpad);
    tk_gemm_kernel<<<TK_T / 128, 256, 0, stream>>>(x, wpad, lin_b, pred);
    tk_score_kernel<<<TK_T / 256, 256, 0, stream>>>(pred, topk_w, score, keys);
    tk_select_kernel<<<TK_B, 256, 0, stream>>>(keys, kth, ties);
    tk_pool_kernel<<<TK_B, 256, 0, stream>>>(pred, score, keys, kth, ties, x_final);
}